// DynamicGraphConv_57664230916671
// MI455X (gfx1250) — compile-verified
//
#include <hip/hip_runtime.h>

// ---------------------------------------------------------------------------
// DynamicGraphConv for MI455X (gfx1250, wave32).
// Re-associated algorithm: U[b,i,h,o] = sum_f x[b,i,f] * w1[h,f,o]  (GEMM 1)
//                          H[b,i,j,h] = relu(rel@w0 + b0)           (GEMM 2)
//                          msg        = adj * (H @ U + x@b1)        (GEMM 3)
//                          out[b,j,o] = bias[o] + sum_i msg
// All GEMMs run on V_WMMA_F32_16X16X4_F32 (exact f32 matrix pipe).
// Workspace: 4MB (U) + 4MB (msg) = 8MB.
// ---------------------------------------------------------------------------

typedef __attribute__((ext_vector_type(2))) float v2f;
typedef __attribute__((ext_vector_type(8))) float v8f;

#define BZ   8
#define NN   64
#define FIN  32
#define FOUT 32
#define FE   16
#define HH   64
#define P    (BZ * NN)     // 512 (b,i) pairs
#define HO   (HH * FOUT)   // 2048

static __device__ __forceinline__ v8f wmma_f32(v2f a, v2f b, v8f c) {
  // D = A(16x4 f32) * B(4x16 f32) + C(16x16 f32); 8-arg VOP3P form:
  // (neg_a, A, neg_b, B, c_mod, C, reuse_a, reuse_b)
  return __builtin_amdgcn_wmma_f32_16x16x4_f32(false, a, false, b, (short)0, c,
                                               false, false);
}

// ---------------------------------------------------------------------------
// Kernel 1: U[p, h*32+o] = sum_f X[p,f] * w1[h, f*32+o]
// GEMM [512,32] x [32,2048]; one 16x16 tile per wave, K=32 -> 8 chained WMMAs.
// A lane layout (ISA 7.12.2, 32-bit A 16x4): lanes 0-15 = (K0,K1), 16-31 = (K2,K3)
// ---------------------------------------------------------------------------
__global__ __launch_bounds__(256) void k_uall(const float* __restrict__ X,
                                              const float* __restrict__ w1,
                                              float* __restrict__ U) {
  const int lane = threadIdx.x & 31;
  const int wave = threadIdx.x >> 5;
  const int tile = blockIdx.x * 8 + wave;  // 0..4095
  const int r = tile >> 7;                 // row-block over P   (0..31)
  const int c = tile & 127;                // col-block over HO  (0..127)
  const int hf = lane >> 4;                // lane half selects K pair
  const int l  = lane & 15;
  const int row = r * 16 + l;              // p index (A rows / D rows)
  const int col = c * 16 + l;              // ho index (B cols / D cols)
  const int h = col >> 5;
  const int o = col & 31;

  v8f acc = {};
#pragma unroll
  for (int k = 0; k < 8; ++k) {
    const int f0 = 4 * k + 2 * hf;
    v2f a, b;
    a.x = X[row * FIN + f0];
    a.y = X[row * FIN + f0 + 1];
    b.x = w1[h * (FIN * FOUT) + f0 * FOUT + o];        // M[f, (h,o)] = permuted w1
    b.y = w1[h * (FIN * FOUT) + (f0 + 1) * FOUT + o];
    acc = wmma_f32(a, b, acc);
  }
#pragma unroll
  for (int v = 0; v < 8; ++v) {
    const int m = v + 8 * hf;              // C/D: VGPR v -> M = v (+8 for hi half)
    U[(r * 16 + m) * HO + col] = acc[v];
  }
}

// ---------------------------------------------------------------------------
// Kernel 2: one workgroup (8 waves) per p=(b,i).
//  Phase 1: sH[j,h] = relu(rel[p] @ w0 + b0)        [64,16]x[16,64], WMMA
//  Phase 2: Mg[j,o] = sH @ sU                       [64,64]x[64,32], WMMA
//  Phase 3: msg[p,j,o] = adj[b,i,j] * (Mg + c[o])
// ---------------------------------------------------------------------------
__global__ __launch_bounds__(256) void k_msg(const float* __restrict__ X,
                                             const float* __restrict__ adj,
                                             const float* __restrict__ rel,
                                             const float* __restrict__ w0,
                                             const float* __restrict__ b0,
                                             const float* __restrict__ b1,
                                             const float* __restrict__ U,
                                             float* __restrict__ msg) {
  __shared__ float sH[NN * HH];     // 16 KB
  __shared__ float sU[HH * FOUT];   //  8 KB
  __shared__ float sC[FOUT];
  __shared__ float sAdj[NN];

  const int p    = blockIdx.x;      // (b,i) flat
  const int tid  = threadIdx.x;
  const int lane = tid & 31;
  const int wave = tid >> 5;
  const int hf   = lane >> 4;
  const int l    = lane & 15;

  // Phase 0: stage U tile, adjacency row, and bias projection c[o] = x @ b1
  for (int q = tid; q < HH * FOUT; q += 256) sU[q] = U[p * HO + q];
  if (tid < NN) sAdj[tid] = adj[p * NN + tid];
  if (tid < FOUT) {
    float acc = 0.f;
    for (int f = 0; f < FIN; ++f) acc += X[p * FIN + f] * b1[f * FOUT + tid];
    sC[tid] = acc;
  }

  // Phase 1: hidden activations, 16 tiles (4x4), 2 per wave, K=16 (4 WMMAs)
  const float* relp = rel + (size_t)p * NN * FE;
#pragma unroll
  for (int t = wave; t < 16; t += 8) {
    const int tj = t >> 2, th = t & 3;
    v8f acc = {};
#pragma unroll
    for (int k = 0; k < 4; ++k) {
      const int f0 = 4 * k + 2 * hf;
      v2f a, b;
      a.x = relp[(tj * 16 + l) * FE + f0];
      a.y = relp[(tj * 16 + l) * FE + f0 + 1];
      b.x = w0[f0 * HH + th * 16 + l];
      b.y = w0[(f0 + 1) * HH + th * 16 + l];
      acc = wmma_f32(a, b, acc);
    }
#pragma unroll
    for (int v = 0; v < 8; ++v) {
      const int j  = tj * 16 + v + 8 * hf;
      const int hc = th * 16 + l;
      float val = acc[v] + b0[hc];
      sH[j * HH + hc] = val > 0.f ? val : 0.f;
    }
  }
  __syncthreads();

  // Phase 2: Mg = sH @ sU, 8 tiles (4x2), one per wave, K=64 (16 WMMAs)
  const int tj = wave >> 1, to = wave & 1;
  v8f acc = {};
#pragma unroll
  for (int k = 0; k < 16; ++k) {
    const int f0 = 4 * k + 2 * hf;
    v2f a, b;
    a.x = sH[(tj * 16 + l) * HH + f0];
    a.y = sH[(tj * 16 + l) * HH + f0 + 1];
    b.x = sU[f0 * FOUT + to * 16 + l];
    b.y = sU[(f0 + 1) * FOUT + to * 16 + l];
    acc = wmma_f32(a, b, acc);
  }

  // Phase 3: bias projection, adjacency scaling, store per-source messages
#pragma unroll
  for (int v = 0; v < 8; ++v) {
    const int j = tj * 16 + v + 8 * hf;
    const int o = to * 16 + l;
    msg[((size_t)p * NN + j) * FOUT + o] = sAdj[j] * (acc[v] + sC[o]);
  }
}

// ---------------------------------------------------------------------------
// Kernel 3: out[b,j,o] = bias[o] + sum_i msg[b,i,j,o]   (deterministic reduce)
// ---------------------------------------------------------------------------
__global__ __launch_bounds__(256) void k_reduce(const float* __restrict__ msg,
                                                const float* __restrict__ bias,
                                                float* __restrict__ out) {
  const int g = blockIdx.x * 256 + threadIdx.x;   // = b*2048 + j*32 + o
  const int o = g & 31;
  const int j = (g >> 5) & 63;
  const int b = g >> 11;
  float acc = bias[o];
  const float* m = msg + (((size_t)b * NN) * NN + j) * FOUT + o;
  for (int i = 0; i < NN; ++i) acc += m[(size_t)i * NN * FOUT];
  out[g] = acc;
}

extern "C" void kernel_launch(void* const* d_in, const int* in_sizes, int n_in,
                              void* d_out, int out_size, void* d_ws, size_t ws_size,
                              hipStream_t stream) {
  const float* X    = (const float*)d_in[0];  // [8,64,32]
  const float* adj  = (const float*)d_in[1];  // [8,64,64]
  const float* rel  = (const float*)d_in[2];  // [8,64,64,16]
  const float* w0   = (const float*)d_in[3];  // [16,64]
  const float* b0   = (const float*)d_in[4];  // [64]
  const float* w1   = (const float*)d_in[5];  // [64,1024]
  const float* b1   = (const float*)d_in[6];  // [1024]
  const float* bias = (const float*)d_in[7];  // [32]
  float* out = (float*)d_out;                 // [8,64,32]

  float* U   = (float*)d_ws;                  // 512*2048 f32 = 4 MB
  float* msg = U + (size_t)P * HO;            // 512*64*32 f32 = 4 MB

  k_uall  <<<P,  256, 0, stream>>>(X, w1, U);
  k_msg   <<<P,  256, 0, stream>>>(X, adj, rel, w0, b0, b1, U, msg);
  k_reduce<<<(P * NN * FOUT) / (NN * 256) * NN / NN, 256, 0, stream>>>(msg, bias, out);
}